// DGCNNBlock_41274635714770
// MI455X (gfx1250) — compile-verified
//
#include <hip/hip_runtime.h>
#include <hip/hip_bf16.h>

#define NPTS 8192
#define BATCH 4
#define KNN_K 20
#define SLOPE 0.2f
#define BN_EPS 1e-5f

typedef float v2f __attribute__((ext_vector_type(2)));
typedef float v8f __attribute__((ext_vector_type(8)));

// D(16x16,f32) = A(16x4,f32) x B(4x16,f32) + C
// 8-arg form: (neg_a, A, neg_b, B, c_mod, C, reuse_a, reuse_b)
__device__ __forceinline__ v8f wmma4(v2f a, v2f b, v8f c) {
    return __builtin_amdgcn_wmma_f32_16x16x4_f32(false, a, false, b, (short)0, c,
                                                 false, false);
}

__device__ __forceinline__ unsigned enc_f32(float x) {
    unsigned u = __float_as_uint(x);
    return ((int)u >= 0) ? (u | 0x80000000u) : ~u;   // order-preserving map
}

// ---------------------------------------------------------------------------
// Pass 0: transpose x (B,C,N) -> f (B,N,C) and squared norms sq (B,N)
// ---------------------------------------------------------------------------
__global__ __launch_bounds__(256) void prep_kernel(const float* __restrict__ x,
                                                   float* __restrict__ f,
                                                   float* __restrict__ sq) {
    int tid = blockIdx.x * blockDim.x + threadIdx.x;   // 0 .. B*N-1
    int b = tid / NPTS;
    int n = tid % NPTS;
    const float* xb = x + (size_t)b * 64 * NPTS;
    float* fr = f + (size_t)tid * 64;
    float acc = 0.f;
#pragma unroll
    for (int c = 0; c < 64; ++c) {
        float v = xb[(size_t)c * NPTS + n];    // coalesced across threads
        acc += v * v;
        fr[c] = v;
    }
    sq[tid] = acc;
}

// ---------------------------------------------------------------------------
// Pass 1: fused distance GEMM (f32 WMMA) + per-row top-20 selection.
// One wave (32 threads) per block; wave owns 32 rows, loops over all 8192
// columns in 16-wide tiles; per column tile: 32 x v_wmma_f32_16x16x4_f32.
// ---------------------------------------------------------------------------
__global__ __launch_bounds__(32) void knn_kernel(const float* __restrict__ f,
                                                 const float* __restrict__ sq,
                                                 int* __restrict__ knn) {
    const int lane = threadIdx.x;
    const int tile = blockIdx.x;                 // B*N/32 tiles
    const int b = tile / (NPTS / 32);
    const int r0 = (tile % (NPTS / 32)) * 32;
    const float* fb = f + (size_t)b * NPTS * 64;
    const float* sqb = sq + (size_t)b * NPTS;
    const int m = lane & 15, lh = lane >> 4, kh = lh * 2;

    // A fragments for the wave's 32 fixed rows (two 16-row subtiles), all K.
    v2f aLo[16], aHi[16];
    {
        const float* pLo = fb + (size_t)(r0 + m) * 64 + kh;
        const float* pHi = pLo + 16 * 64;
#pragma unroll
        for (int t = 0; t < 16; ++t) {
            aLo[t] = *(const v2f*)(pLo + 4 * t);
            aHi[t] = *(const v2f*)(pHi + 4 * t);
        }
    }
    float sqLo[8], sqHi[8];
#pragma unroll
    for (int v = 0; v < 8; ++v) {
        sqLo[v] = sqb[r0 + v + 8 * lh];
        sqHi[v] = sqb[r0 + 16 + v + 8 * lh];
    }

    // Per-lane (one row per lane) sorted top-20 smallest distances.
    float dl[KNN_K];
    int il[KNN_K];
#pragma unroll
    for (int q = 0; q < KNN_K; ++q) { dl[q] = 3.4e38f; il[q] = 0; }

    __shared__ float tileD[2][16][17];           // padded: conflict-free rows

    for (int c0 = 0; c0 < NPTS; c0 += 16) {
        v8f cLo = {0, 0, 0, 0, 0, 0, 0, 0};
        v8f cHi = {0, 0, 0, 0, 0, 0, 0, 0};
        const float* pc = fb + (size_t)(c0 + m) * 64 + kh;   // B fragment base
#pragma unroll
        for (int t = 0; t < 16; ++t) {
            v2f bf = *(const v2f*)(pc + 4 * t);
            cLo = wmma4(aLo[t], bf, cLo);
            cHi = wmma4(aHi[t], bf, cHi);
        }
        float sqc = sqb[c0 + m];
#pragma unroll
        for (int v = 0; v < 8; ++v) {
            tileD[0][v + 8 * lh][m] = sqLo[v] + sqc - 2.0f * cLo[v];
            tileD[1][v + 8 * lh][m] = sqHi[v] + sqc - 2.0f * cHi[v];
        }
        __syncthreads();
        float cand[16];
#pragma unroll
        for (int j = 0; j < 16; ++j) cand[j] = tileD[lh][m][j];
        float mn = cand[0];
#pragma unroll
        for (int j = 1; j < 16; ++j) mn = fminf(mn, cand[j]);
        if (mn < dl[KNN_K - 1]) {                // prefilter (divergence cut)
#pragma unroll
            for (int j = 0; j < 16; ++j) {
                float d = cand[j];
                if (d < dl[KNN_K - 1]) {
                    int ci = c0 + j;
#pragma unroll
                    for (int q = KNN_K - 1; q > 0; --q) {
                        bool lt_prev = d < dl[q - 1];
                        bool lt_cur = d < dl[q];
                        float nd = lt_prev ? dl[q - 1] : (lt_cur ? d : dl[q]);
                        int ni = lt_prev ? il[q - 1] : (lt_cur ? ci : il[q]);
                        dl[q] = nd;
                        il[q] = ni;
                    }
                    if (d < dl[0]) { dl[0] = d; il[0] = ci; }
                }
            }
        }
        __syncthreads();
    }
    int* op = knn + ((size_t)b * NPTS + r0 + lane) * KNN_K;
#pragma unroll
    for (int q = 0; q < KNN_K; ++q) op[q] = il[q];
}

// ---------------------------------------------------------------------------
// Conv passes. Rows = (b,n,k) edge samples, M-tiled by 32 per wave.
// edge feature trick: h1 = f_ne @ W1a^T + f_ctr @ (W1b - W1a)^T
// MODE 0: conv1 -> BN1 stats.  MODE 1: conv1+bn1+lrelu -> conv2 -> BN2 stats.
// MODE 2: full chain + lrelu + encoded atomicMax over k into enc buffer.
// stats layout (floats): [0]sum1 [64]sumsq1 [128]scale1 [192]shift1
//                        [256]sum2 [320]sumsq2 [384]scale2 [448]shift2
// ---------------------------------------------------------------------------
template <int MODE>
__global__ __launch_bounds__(128) void conv_kernel(const float* __restrict__ f,
                                                   const int* __restrict__ knn,
                                                   const float* __restrict__ W1,
                                                   const float* __restrict__ W2,
                                                   float* __restrict__ stats,
                                                   unsigned* __restrict__ enc) {
    const int lane = threadIdx.x & 31;
    const int wid = threadIdx.x >> 5;
    const int tileIdx = blockIdx.x * 4 + wid;        // 20480 tiles total
    const int Rbase = tileIdx * 32;
    const int b = Rbase / (NPTS * KNN_K);
    const int rb = Rbase % (NPTS * KNN_K);           // row within batch
    const float* fb = f + (size_t)b * NPTS * 64;
    const int m = lane & 15, lh = lane >> 4, kh = lh * 2;

    __shared__ float sAcc[128];
    __shared__ float hs[4][32][66];                  // per-wave hn1 tile, padded

    if (MODE != 2) {
        sAcc[threadIdx.x] = 0.f;
        __syncthreads();
    }

    // A fragments: neighbor rows (gathered via knn) and center rows.
    const int nLo = (rb + m) / KNN_K;
    const int nHi = (rb + 16 + m) / KNN_K;
    const int jLo = knn[(size_t)b * NPTS * KNN_K + rb + m];
    const int jHi = knn[(size_t)b * NPTS * KNN_K + rb + 16 + m];
    v2f aneL[16], aneH[16], actL[16], actH[16];
    {
        const float* pNL = fb + (size_t)jLo * 64 + kh;
        const float* pNH = fb + (size_t)jHi * 64 + kh;
        const float* pCL = fb + (size_t)nLo * 64 + kh;
        const float* pCH = fb + (size_t)nHi * 64 + kh;
#pragma unroll
        for (int t = 0; t < 16; ++t) {
            aneL[t] = *(const v2f*)(pNL + 4 * t);
            aneH[t] = *(const v2f*)(pNH + 4 * t);
            actL[t] = *(const v2f*)(pCL + 4 * t);
            actH[t] = *(const v2f*)(pCH + 4 * t);
        }
    }

    // ---- conv1: 4 output-channel tiles of 16 ----
    for (int q = 0; q < 4; ++q) {
        v8f cL = {0, 0, 0, 0, 0, 0, 0, 0};
        v8f cH = {0, 0, 0, 0, 0, 0, 0, 0};
        const float* wrow = W1 + (size_t)(q * 16 + m) * 128 + kh;
#pragma unroll
        for (int t = 0; t < 16; ++t) {
            v2f ba = *(const v2f*)(wrow + 4 * t);         // W1a^T fragment
            v2f bb = *(const v2f*)(wrow + 64 + 4 * t);    // W1b^T fragment
            v2f bc = bb - ba;                             // (W1b-W1a)^T
            cL = wmma4(aneL[t], ba, cL);
            cL = wmma4(actL[t], bc, cL);
            cH = wmma4(aneH[t], ba, cH);
            cH = wmma4(actH[t], bc, cH);
        }
        if (MODE == 0) {
            float s = 0.f, s2 = 0.f;
#pragma unroll
            for (int v = 0; v < 8; ++v) {
                s += cL[v] + cH[v];
                s2 += cL[v] * cL[v] + cH[v] * cH[v];
            }
            atomicAdd(&sAcc[q * 16 + m], s);
            atomicAdd(&sAcc[64 + q * 16 + m], s2);
        } else {
            int o = q * 16 + m;
            float sc = stats[128 + o];
            float sh = stats[192 + o];
#pragma unroll
            for (int v = 0; v < 8; ++v) {
                float a = cL[v] * sc + sh;
                a = (a >= 0.f) ? a : SLOPE * a;
                hs[wid][v + 8 * lh][o] = a;
                float a2 = cH[v] * sc + sh;
                a2 = (a2 >= 0.f) ? a2 : SLOPE * a2;
                hs[wid][16 + v + 8 * lh][o] = a2;
            }
        }
    }
    __syncthreads();

    // ---- conv2 on normalized h1 (MODE 1/2) ----
    if (MODE >= 1) {
        for (int q = 0; q < 4; ++q) {
            v8f cL = {0, 0, 0, 0, 0, 0, 0, 0};
            v8f cH = {0, 0, 0, 0, 0, 0, 0, 0};
            const float* w2row = W2 + (size_t)(q * 16 + m) * 64 + kh;
#pragma unroll
            for (int t = 0; t < 16; ++t) {
                v2f bb = *(const v2f*)(w2row + 4 * t);
                v2f aL = *(const v2f*)(&hs[wid][m][4 * t + kh]);
                v2f aH = *(const v2f*)(&hs[wid][16 + m][4 * t + kh]);
                cL = wmma4(aL, bb, cL);
                cH = wmma4(aH, bb, cH);
            }
            if (MODE == 1) {
                float s = 0.f, s2 = 0.f;
#pragma unroll
                for (int v = 0; v < 8; ++v) {
                    s += cL[v] + cH[v];
                    s2 += cL[v] * cL[v] + cH[v] * cH[v];
                }
                atomicAdd(&sAcc[q * 16 + m], s);
                atomicAdd(&sAcc[64 + q * 16 + m], s2);
            } else {   // MODE 2: normalize, lrelu, max over k via encoded u32
                int o = q * 16 + m;
                float sc = stats[384 + o];
                float sh = stats[448 + o];
                unsigned* eb = enc + ((size_t)b * 64 + o) * NPTS;
#pragma unroll
                for (int v = 0; v < 8; ++v) {
                    int lr = v + 8 * lh;
                    float a = cL[v] * sc + sh;
                    a = (a >= 0.f) ? a : SLOPE * a;
                    atomicMax(&eb[(rb + lr) / KNN_K], enc_f32(a));
                    float a2 = cH[v] * sc + sh;
                    a2 = (a2 >= 0.f) ? a2 : SLOPE * a2;
                    atomicMax(&eb[(rb + 16 + lr) / KNN_K], enc_f32(a2));
                }
            }
        }
    }

    if (MODE != 2) {
        __syncthreads();
        atomicAdd(&stats[(MODE == 0 ? 0 : 256) + threadIdx.x], sAcc[threadIdx.x]);
    }
}

// ---------------------------------------------------------------------------
// BN finalize: mean/var -> fused scale/shift
// ---------------------------------------------------------------------------
__global__ void finalize_kernel(const float* __restrict__ g,
                                const float* __restrict__ bta,
                                float* __restrict__ stats, int off) {
    int o = threadIdx.x;
    if (o >= 64) return;
    const float cnt = (float)BATCH * (float)NPTS * (float)KNN_K;
    float mu = stats[off + o] / cnt;
    float var = stats[off + 64 + o] / cnt - mu * mu;
    float rs = rsqrtf(var + BN_EPS);
    float sc = g[o] * rs;
    stats[off + 128 + o] = sc;
    stats[off + 192 + o] = bta[o] - mu * sc;
}

// ---------------------------------------------------------------------------
// Decode order-preserving u32 back to f32
// ---------------------------------------------------------------------------
__global__ __launch_bounds__(256) void decode_kernel(const unsigned* __restrict__ enc,
                                                     float* __restrict__ out, int n) {
    int i = blockIdx.x * blockDim.x + threadIdx.x;
    if (i < n) {
        unsigned u = enc[i];
        unsigned v = (u & 0x80000000u) ? (u & 0x7fffffffu) : ~u;
        out[i] = __uint_as_float(v);
    }
}

// ---------------------------------------------------------------------------
extern "C" void kernel_launch(void* const* d_in, const int* in_sizes, int n_in,
                              void* d_out, int out_size, void* d_ws, size_t ws_size,
                              hipStream_t stream) {
    const float* x = (const float*)d_in[0];   // (4,64,8192)
    const float* W1 = (const float*)d_in[1];  // (64,128)
    const float* g1 = (const float*)d_in[2];
    const float* b1 = (const float*)d_in[3];
    const float* W2 = (const float*)d_in[4];  // (64,64)
    const float* g2 = (const float*)d_in[5];
    const float* b2 = (const float*)d_in[6];

    char* ws = (char*)d_ws;
    float* f = (float*)(ws);                       // 8 MB : (B,N,64)
    float* sq = (float*)(ws + 8388608);            // 128 KB
    int* knn = (int*)(ws + 8519680);               // 2.5 MB : (B,N,20)
    float* stats = (float*)(ws + 11141120);        // 2 KB
    unsigned* enc = (unsigned*)(ws + 11143168);    // 8 MB encoded output
    float* out = (float*)d_out;

    hipMemsetAsync(stats, 0, 2048, stream);
    hipMemsetAsync(enc, 0, (size_t)BATCH * 64 * NPTS * 4, stream);

    prep_kernel<<<(BATCH * NPTS) / 256, 256, 0, stream>>>(x, f, sq);
    knn_kernel<<<(BATCH * NPTS) / 32, 32, 0, stream>>>(f, sq, knn);

    const int nTiles = BATCH * NPTS * KNN_K / 32;  // 20480
    conv_kernel<0><<<nTiles / 4, 128, 0, stream>>>(f, knn, W1, W2, stats, enc);
    finalize_kernel<<<1, 64, 0, stream>>>(g1, b1, stats, 0);
    conv_kernel<1><<<nTiles / 4, 128, 0, stream>>>(f, knn, W1, W2, stats, enc);
    finalize_kernel<<<1, 64, 0, stream>>>(g2, b2, stats, 256);
    conv_kernel<2><<<nTiles / 4, 128, 0, stream>>>(f, knn, W1, W2, stats, enc);

    const int outN = BATCH * 64 * NPTS;
    decode_kernel<<<(outN + 255) / 256, 256, 0, stream>>>(enc, out, outN);
}